// NGCF_58841051955230
// MI455X (gfx1250) — compile-verified
//
#include <hip/hip_runtime.h>

#define N_USER  50000
#define N_ITEM  100000
#define N_NODES (N_USER + N_ITEM)
#define EMB     64
#define N_LAYERS 3
#define NNZ     2400000
#define BATCH   4096
#define WLD     66   // padded LDS row stride for transposed weights (even -> 8B aligned pairs)

typedef __attribute__((ext_vector_type(2))) float v2f;
typedef __attribute__((ext_vector_type(8))) float v8f;

// ---------------------------------------------------------------------------
// Init: ego[:,0:64] = concat(user_emb,item_emb); all_embs[:,0:64] = same.
// One float4 per thread; grid = N_NODES*16 float4 units.
// ---------------------------------------------------------------------------
__global__ __launch_bounds__(256) void ngcf_init_kernel(
    const float* __restrict__ uemb, const float* __restrict__ iemb,
    float* __restrict__ ego, float* __restrict__ allemb)
{
    unsigned i    = blockIdx.x * 256u + threadIdx.x;   // float4 unit index
    unsigned node = i >> 4;
    unsigned q    = i & 15;
    float4 v;
    if (node < N_USER) v = ((const float4*)uemb)[(size_t)node * 16 + q];
    else               v = ((const float4*)iemb)[(size_t)(node - N_USER) * 16 + q];
    ((float4*)ego)[i] = v;
    // all_embs row = 256 floats = 64 float4; first 16 float4 = slice 0
    ((float4*)allemb)[(size_t)node * 64 + q] = v;
}

// ---------------------------------------------------------------------------
// Zero `side` before atomic accumulation (d_ws is poisoned, not zeroed).
// ---------------------------------------------------------------------------
__global__ __launch_bounds__(256) void ngcf_zero_kernel(float4* __restrict__ p)
{
    unsigned i = blockIdx.x * 256u + threadIdx.x;      // exactly N_NODES*16 units
    p[i] = make_float4(0.f, 0.f, 0.f, 0.f);
}

// ---------------------------------------------------------------------------
// SpMM: side[row] += val * ego[col] for each edge. 16 lanes per edge, each
// lane owns 4 floats (float4 gather + 4 global_atomic_add_f32). Both tables
// (38.4 MB each) are L2-resident on MI455X's 192 MB L2.
// ---------------------------------------------------------------------------
__global__ __launch_bounds__(256) void ngcf_spmm_kernel(
    const int* __restrict__ arow, const int* __restrict__ acol,
    const float* __restrict__ aval,
    const float* __restrict__ ego, float* __restrict__ side)
{
    unsigned tid = blockIdx.x * 256u + threadIdx.x;    // NNZ*16 threads
    unsigned e = tid >> 4;
    unsigned j = (tid & 15u) << 2;                     // 0..60 step 4
    int   r = arow[e];
    int   c = acol[e];
    float v = aval[e];
    float4 x = *(const float4*)(ego + (size_t)c * EMB + j);
    float* dst = side + (size_t)r * EMB + j;
    atomicAdd(dst + 0, v * x.x);
    atomicAdd(dst + 1, v * x.y);
    atomicAdd(dst + 2, v * x.z);
    atomicAdd(dst + 3, v * x.w);
}

// ---------------------------------------------------------------------------
// Fused dual GEMM via V_WMMA_F32_16X16X4_F32:
//   pre = side @ Wgc + (ego*side) @ Wbi + bgc + bbi
// One wave computes a 16x64 output tile (4 16x16 tiles), K=64 in 16 steps of 4.
// Both GEMMs accumulate into the same C (bias preloaded in C).
// Weights are stored TRANSPOSED in LDS (sW[n*WLD + k]) so each B-fragment
// {W[ka][n], W[ka+1][n]} is one 8B-aligned ds_load_b64 straight into an
// even VGPR pair -- no repacking moves before the WMMA.
// A-fragment (16x4 f32): lane l -> M=l&15, v0=K(2*(l>>4)), v1=K+1 (one v2f load)
// D/C: lane l, reg r -> D[r + 8*(l>>4)][l&15] within the 16x16 tile.
// `pre` aliases `side` (in-place): each wave reads only its own 16 rows fully
// before storing them; no __restrict__ on side/pre so ordering is preserved.
// ---------------------------------------------------------------------------
__global__ __launch_bounds__(128) void ngcf_gemm_kernel(
    const float* side, const float* ego,
    const float* __restrict__ Wgc, const float* __restrict__ bgc,
    const float* __restrict__ Wbi, const float* __restrict__ bbi,
    float* pre)
{
    __shared__ float sWg[EMB * WLD];   // ~16.5 KB, transposed: sWg[n*WLD + k]
    __shared__ float sWb[EMB * WLD];
    {
        unsigned t = threadIdx.x;
        const float4* g4 = (const float4*)Wgc;
        const float4* b4 = (const float4*)Wbi;
        #pragma unroll
        for (int i = 0; i < 8; ++i) {              // 1024 float4 total / 128 threads
            unsigned idx = t + i * 128u;           // float4 unit in row-major W
            unsigned k   = idx >> 4;               // K row 0..63
            unsigned n0  = (idx & 15u) * 4u;       // N column base
            float4 vg = g4[idx];
            float4 vb = b4[idx];
            sWg[(n0 + 0) * WLD + k] = vg.x;
            sWg[(n0 + 1) * WLD + k] = vg.y;
            sWg[(n0 + 2) * WLD + k] = vg.z;
            sWg[(n0 + 3) * WLD + k] = vg.w;
            sWb[(n0 + 0) * WLD + k] = vb.x;
            sWb[(n0 + 1) * WLD + k] = vb.y;
            sWb[(n0 + 2) * WLD + k] = vb.z;
            sWb[(n0 + 3) * WLD + k] = vb.w;
        }
    }
    __syncthreads();

    const unsigned wave    = threadIdx.x >> 5;
    const unsigned lane    = threadIdx.x & 31u;
    const unsigned rowTile = blockIdx.x * 4u + wave;
    if (rowTile >= (N_NODES / 16)) return;             // wave-uniform: EXEC all-1s below
    const unsigned row0 = rowTile * 16u;
    const unsigned m = lane & 15u;
    const unsigned g = lane >> 4;

    // C preloaded with bias: bias depends only on column n = t*16 + m.
    v8f acc[4];
    #pragma unroll
    for (int t = 0; t < 4; ++t) {
        float b = bgc[t * 16 + m] + bbi[t * 16 + m];
        #pragma unroll
        for (int r = 0; r < 8; ++r) acc[t][r] = b;
    }

    const float* Srow = side + (size_t)(row0 + m) * EMB;
    const float* Erow = ego  + (size_t)(row0 + m) * EMB;

    // Per-tile transposed-weight row base (n = t*16 + m)
    const float* wgRow[4];
    const float* wbRow[4];
    #pragma unroll
    for (int t = 0; t < 4; ++t) {
        wgRow[t] = sWg + (size_t)(t * 16 + m) * WLD;
        wbRow[t] = sWb + (size_t)(t * 16 + m) * WLD;
    }

    #pragma unroll
    for (int k0 = 0; k0 < EMB; k0 += 4) {
        const int ka = k0 + 2 * (int)g;                // this lane's K pair (even)
        v2f s2 = *(const v2f*)(Srow + ka);             // A_gc fragment
        v2f e2 = *(const v2f*)(Erow + ka);
        v2f p2 = s2 * e2;                              // A_bi fragment (ego*side)
        #pragma unroll
        for (int t = 0; t < 4; ++t) {
            v2f bg = *(const v2f*)(wgRow[t] + ka);     // single ds_load_b64
            v2f bb = *(const v2f*)(wbRow[t] + ka);     // single ds_load_b64
            acc[t] = __builtin_amdgcn_wmma_f32_16x16x4_f32(
                false, s2, false, bg, (short)0, acc[t], false, false);
            acc[t] = __builtin_amdgcn_wmma_f32_16x16x4_f32(
                false, p2, false, bb, (short)0, acc[t], false, false);
        }
    }

    #pragma unroll
    for (int t = 0; t < 4; ++t) {
        #pragma unroll
        for (int r = 0; r < 8; ++r) {
            pre[(size_t)(row0 + 8 * g + r) * EMB + t * 16 + m] = acc[t][r];
        }
    }
}

// ---------------------------------------------------------------------------
// Post: ego = leaky_relu(pre, 0.2); all_embs[:, 64*(k+1):64*(k+2)] = ego/||ego||.
// One wave32 per node (2 floats/lane), sum-of-squares via shfl_xor reduction.
// ---------------------------------------------------------------------------
__global__ __launch_bounds__(256) void ngcf_post_kernel(
    const float* __restrict__ pre, float* __restrict__ ego,
    float* __restrict__ allemb, int layer)
{
    unsigned node = (blockIdx.x * 256u + threadIdx.x) >> 5;   // one wave per node
    unsigned lane = threadIdx.x & 31u;
    if (node >= N_NODES) return;

    v2f x = *(const v2f*)(pre + (size_t)node * EMB + lane * 2);
    v2f y;
    y.x = (x.x >= 0.f) ? x.x : 0.2f * x.x;
    y.y = (x.y >= 0.f) ? x.y : 0.2f * x.y;

    float ss = y.x * y.x + y.y * y.y;
    #pragma unroll
    for (int off = 16; off > 0; off >>= 1) ss += __shfl_xor(ss, off, 32);
    float inv = 1.0f / fmaxf(sqrtf(ss), 1e-12f);

    *(v2f*)(ego + (size_t)node * EMB + lane * 2) = y;         // unnormalized -> next layer
    float* dst = allemb + (size_t)node * 256 + (size_t)(layer + 1) * EMB + lane * 2;
    dst[0] = y.x * inv;
    dst[1] = y.y * inv;
}

// ---------------------------------------------------------------------------
// Final gather: out = [allemb[users]; allemb[N_USER+pos]; allemb[N_USER+neg]],
// 256 floats per row, one float4 per thread (coalesced).
// ---------------------------------------------------------------------------
__global__ __launch_bounds__(256) void ngcf_gather_kernel(
    const float* __restrict__ allemb,
    const int* __restrict__ users, const int* __restrict__ pos,
    const int* __restrict__ neg, float* __restrict__ out)
{
    unsigned tid = blockIdx.x * 256u + threadIdx.x;   // 3*BATCH*64 float4 units
    unsigned seg = tid / (BATCH * 64);
    unsigned rem = tid - seg * (BATCH * 64);
    unsigned b = rem >> 6;
    unsigned q = rem & 63u;
    int node;
    if (seg == 0)      node = users[b];
    else if (seg == 1) node = N_USER + pos[b];
    else               node = N_USER + neg[b];
    ((float4*)out)[tid] = ((const float4*)(allemb + (size_t)node * 256))[q];
}

// ---------------------------------------------------------------------------
extern "C" void kernel_launch(void* const* d_in, const int* in_sizes, int n_in,
                              void* d_out, int out_size, void* d_ws, size_t ws_size,
                              hipStream_t stream)
{
    const int*   users = (const int*)d_in[0];
    const int*   pos   = (const int*)d_in[1];
    const int*   neg   = (const int*)d_in[2];
    const int*   arow  = (const int*)d_in[3];
    const int*   acol  = (const int*)d_in[4];
    const float* aval  = (const float*)d_in[5];
    const float* uemb  = (const float*)d_in[6];
    const float* iemb  = (const float*)d_in[7];
    const float* Wgc   = (const float*)d_in[8];
    const float* bgc   = (const float*)d_in[9];
    const float* Wbi   = (const float*)d_in[10];
    const float* bbi   = (const float*)d_in[11];
    float* out = (float*)d_out;

    // Workspace layout (floats): all_embs[N_NODES*256] | ego[N_NODES*64] | side[N_NODES*64]
    float* allemb = (float*)d_ws;
    float* ego    = allemb + (size_t)N_NODES * 256;
    float* side   = ego    + (size_t)N_NODES * EMB;

    ngcf_init_kernel<<<(N_NODES * 16) / 256, 256, 0, stream>>>(uemb, iemb, ego, allemb);

    for (int k = 0; k < N_LAYERS; ++k) {
        ngcf_zero_kernel<<<(N_NODES * 16) / 256, 256, 0, stream>>>((float4*)side);
        ngcf_spmm_kernel<<<(NNZ * 16) / 256, 256, 0, stream>>>(arow, acol, aval, ego, side);
        ngcf_gemm_kernel<<<(N_NODES / 16 + 3) / 4, 128, 0, stream>>>(
            side, ego,
            Wgc + (size_t)k * EMB * EMB, bgc + (size_t)k * EMB,
            Wbi + (size_t)k * EMB * EMB, bbi + (size_t)k * EMB,
            side /* pre, in-place */);
        ngcf_post_kernel<<<(N_NODES + 7) / 8, 256, 0, stream>>>(side, ego, allemb, k);
    }

    ngcf_gather_kernel<<<(3 * BATCH * 64) / 256, 256, 0, stream>>>(allemb, users, pos, neg, out);
}